// SimpleGNN_90718299226218
// MI455X (gfx1250) — compile-verified
//
#include <hip/hip_runtime.h>
#include <hip/hip_bf16.h>

typedef float v2f __attribute__((ext_vector_type(2)));
typedef float v8f __attribute__((ext_vector_type(8)));

#define D 128            // feature dim (all layers)
#define NGRAPH 512
#define NCLASS 2
#define WSTRIDE 136      // LDS row stride (floats): keeps lo/hi lane-half bank
                         // sets disjoint (2*136 mod 64 == 16) -> conflict-free

// ---------------------------------------------------------------------------
// GEMM  H[M x 128] = X[M x 128] @ W[128 x 128]  via V_WMMA_F32_16X16X4_F32.
// W is staged once per workgroup into LDS (padded stride); each of the 8
// waves computes a 16x128 slab: 8 v8f accumulators, A-fragment reused across
// the 8 N-tiles. M must be a multiple of 16 (50000 and 512 both are).
// ---------------------------------------------------------------------------
__global__ void __launch_bounds__(256)
gemm_f32_wmma(const float* __restrict__ X, const float* __restrict__ W,
              float* __restrict__ H, int M) {
    __shared__ float Wlds[D * WSTRIDE];            // ~69.6 KB

    // Cooperative fill: float4 chunks never cross a 128-float row boundary.
    const int tid = threadIdx.x;
    for (int i = tid * 4; i < D * D; i += 256 * 4) {
        const int r = i >> 7, c = i & (D - 1);
        *(float4*)(&Wlds[r * WSTRIDE + c]) = *(const float4*)(W + i);
    }
    __syncthreads();                                // all waves participate

    const int wave = (blockIdx.x * blockDim.x + threadIdx.x) >> 5;
    const int lane = threadIdx.x & 31;
    const int row0 = wave * 16;
    if (row0 >= M) return;                          // wave-uniform: EXEC all-1s

    const int lr = lane & 15;
    const int ko = (lane >= 16) ? 2 : 0;            // hi half carries K+2/K+3
    const bool hi = lane >= 16;

    v8f acc[8];
    #pragma unroll
    for (int t = 0; t < 8; ++t) acc[t] = (v8f){0.f,0.f,0.f,0.f,0.f,0.f,0.f,0.f};

    const float* __restrict__ xrow = X + (size_t)(row0 + lr) * D;

    for (int k = 0; k < D; k += 4) {
        // A fragment: contiguous (K, K+1) pair, 8B aligned -> one b64 load.
        const v2f a = *(const v2f*)(xrow + k + ko);
        const float* __restrict__ wr = &Wlds[(k + ko) * WSTRIDE];
        #pragma unroll
        for (int t = 0; t < 8; ++t) {
            const int c = t * 16 + lr;
            v2f b;
            b.x = wr[c];                            // K+ko   row
            b.y = wr[WSTRIDE + c];                  // K+ko+1 row
            acc[t] = __builtin_amdgcn_wmma_f32_16x16x4_f32(
                false, a, false, b, (short)0, acc[t], false, false);
        }
    }

    // C/D layout: VGPR j -> M = j (lanes 0-15) or j+8 (lanes 16-31), N = lane%16
    #pragma unroll
    for (int t = 0; t < 8; ++t) {
        #pragma unroll
        for (int j = 0; j < 8; ++j) {
            const int r = row0 + j + (hi ? 8 : 0);
            H[(size_t)r * D + t * 16 + lr] = acc[t][j];
        }
    }
}

// ---------------------------------------------------------------------------
// Degree / normalization helpers
// ---------------------------------------------------------------------------
__global__ void init_deg(float* deg, int N) {
    int i = blockIdx.x * blockDim.x + threadIdx.x;
    if (i < N) deg[i] = 1.0f;              // self-loop
}

__global__ void count_deg(const int* __restrict__ col, float* deg, int E) {
    int e = blockIdx.x * blockDim.x + threadIdx.x;
    if (e < E) atomicAdd(&deg[col[e]], 1.0f);
}

__global__ void make_dinv(float* deg, int N) {
    int i = blockIdx.x * blockDim.x + threadIdx.x;
    if (i < N) {
        float d = deg[i];
        deg[i] = d > 0.f ? rsqrtf(d) : 0.f;
    }
}

// AGG[i,:] = H[i,:] * dinv[i]^2   (self-loop contribution, also zero-inits AGG)
__global__ void self_loop_init(const float* __restrict__ H,
                               const float* __restrict__ dinv,
                               float* __restrict__ AGG, int N) {
    int idx = blockIdx.x * blockDim.x + threadIdx.x;
    if (idx < N * D) {
        int i = idx >> 7;                  // / 128
        float s = dinv[i];
        AGG[idx] = H[idx] * s * s;
    }
}

// One wave per edge, each lane moves a float4 (whole 512B row per wave).
__global__ void __launch_bounds__(256)
scatter_edges(const int* __restrict__ row, const int* __restrict__ col,
              const float* __restrict__ dinv, const float* __restrict__ H,
              float* __restrict__ AGG, int E) {
    const int e    = (blockIdx.x * blockDim.x + threadIdx.x) >> 5;
    const int lane = threadIdx.x & 31;
    if (e >= E) return;
    const int r = row[e];
    const int c = col[e];
    const float norm = dinv[r] * dinv[c];
    const float4 v = ((const float4*)(H + (size_t)r * D))[lane];
    float* o = AGG + (size_t)c * D + lane * 4;
    atomicAdd(o + 0, v.x * norm);
    atomicAdd(o + 1, v.y * norm);
    atomicAdd(o + 2, v.z * norm);
    atomicAdd(o + 3, v.w * norm);
}

// Y[i,f] = relu(Y[i,f] + b[f])   in-place
__global__ void bias_relu(float* Y, const float* __restrict__ b, int N) {
    int idx = blockIdx.x * blockDim.x + threadIdx.x;
    if (idx < N * D) {
        float v = Y[idx] + b[idx & (D - 1)];
        Y[idx] = v > 0.f ? v : 0.f;
    }
}

// ---------------------------------------------------------------------------
// Global mean pool
// ---------------------------------------------------------------------------
__global__ void pool_clear(float* pool, float* cnt) {
    int i = blockIdx.x * blockDim.x + threadIdx.x;
    if (i < NGRAPH * D) pool[i] = 0.f;
    if (i < NGRAPH) cnt[i] = 0.f;
}

__global__ void __launch_bounds__(256)
pool_sum(const float* __restrict__ X, const int* __restrict__ batch,
         float* pool, int N) {
    const int i    = (blockIdx.x * blockDim.x + threadIdx.x) >> 5;
    const int lane = threadIdx.x & 31;
    if (i >= N) return;
    const int g = batch[i];
    const float4 v = ((const float4*)(X + (size_t)i * D))[lane];
    float* o = pool + (size_t)g * D + lane * 4;
    atomicAdd(o + 0, v.x);
    atomicAdd(o + 1, v.y);
    atomicAdd(o + 2, v.z);
    atomicAdd(o + 3, v.w);
}

__global__ void pool_count(const int* __restrict__ batch, float* cnt, int N) {
    int i = blockIdx.x * blockDim.x + threadIdx.x;
    if (i < N) atomicAdd(&cnt[batch[i]], 1.0f);
}

__global__ void pool_mean(float* pool, const float* __restrict__ cnt) {
    int idx = blockIdx.x * blockDim.x + threadIdx.x;
    if (idx < NGRAPH * D) {
        float c = cnt[idx >> 7];
        pool[idx] /= (c > 1.f ? c : 1.f);
    }
}

// out[g, 0..1] = g1[g,:] @ lin2_W + lin2_b
__global__ void final_lin(const float* __restrict__ g1, const float* __restrict__ W2,
                          const float* __restrict__ b2, float* __restrict__ out) {
    int g = blockIdx.x * blockDim.x + threadIdx.x;
    if (g >= NGRAPH) return;
    float s0 = 0.f, s1 = 0.f;
    const float* r = g1 + (size_t)g * D;
    #pragma unroll 4
    for (int f = 0; f < D; ++f) {
        float v = r[f];
        s0 += v * W2[f * NCLASS + 0];
        s1 += v * W2[f * NCLASS + 1];
    }
    out[g * NCLASS + 0] = s0 + b2[0];
    out[g * NCLASS + 1] = s1 + b2[1];
}

// ---------------------------------------------------------------------------
extern "C" void kernel_launch(void* const* d_in, const int* in_sizes, int n_in,
                              void* d_out, int out_size, void* d_ws, size_t ws_size,
                              hipStream_t stream) {
    const float* x      = (const float*)d_in[0];
    const int*   eidx   = (const int*)d_in[1];
    const int*   batch  = (const int*)d_in[2];
    const float* W1     = (const float*)d_in[3];
    const float* b1     = (const float*)d_in[4];
    const float* W2     = (const float*)d_in[5];
    const float* b2     = (const float*)d_in[6];
    const float* lin1W  = (const float*)d_in[7];
    const float* lin1b  = (const float*)d_in[8];
    const float* lin2W  = (const float*)d_in[9];
    const float* lin2b  = (const float*)d_in[10];
    float* out = (float*)d_out;

    const int N = in_sizes[0] / D;       // 50000
    const int E = in_sizes[1] / 2;       // 1600000
    const int* erow = eidx;              // sources
    const int* ecol = eidx + E;          // targets

    // workspace carve-up (all offsets 256B aligned)
    char* ws = (char*)d_ws;
    size_t off = 0;
    float* H    = (float*)(ws + off); off += (size_t)N * D * sizeof(float);
    float* AGG  = (float*)(ws + off); off += (size_t)N * D * sizeof(float);
    float* dinv = (float*)(ws + off); off += ((size_t)N * sizeof(float) + 255) & ~255ull;
    float* pool = (float*)(ws + off); off += (size_t)NGRAPH * D * sizeof(float);
    float* cnt  = (float*)(ws + off); off += 256;
    float* g1   = (float*)(ws + off); off += (size_t)NGRAPH * D * sizeof(float);
    (void)ws_size; (void)n_in; (void)out_size;

    const int T = 256;
    const int gN     = (N + T - 1) / T;
    const int gE     = (E + T - 1) / T;
    const int gNF    = (N * D + T - 1) / T;           // node*feature grid
    const int gEwave = (E + 7) / 8;                    // 1 wave / edge, 8 waves / block
    const int gNwave = (N + 7) / 8;
    const int gGemmN = ((N / 16) + 7) / 8;             // waves of 16 rows
    const int gGemmG = ((NGRAPH / 16) + 7) / 8;

    // normalization
    init_deg  <<<gN, T, 0, stream>>>(dinv, N);
    count_deg <<<gE, T, 0, stream>>>(ecol, dinv, E);
    make_dinv <<<gN, T, 0, stream>>>(dinv, N);

    // ---- layer 1 ----
    gemm_f32_wmma <<<gGemmN, T, 0, stream>>>(x, W1, H, N);
    self_loop_init<<<gNF, T, 0, stream>>>(H, dinv, AGG, N);
    scatter_edges <<<gEwave, T, 0, stream>>>(erow, ecol, dinv, H, AGG, E);
    bias_relu     <<<gNF, T, 0, stream>>>(AGG, b1, N);

    // ---- layer 2 ----
    gemm_f32_wmma <<<gGemmN, T, 0, stream>>>(AGG, W2, H, N);
    self_loop_init<<<gNF, T, 0, stream>>>(H, dinv, AGG, N);
    scatter_edges <<<gEwave, T, 0, stream>>>(erow, ecol, dinv, H, AGG, E);
    bias_relu     <<<gNF, T, 0, stream>>>(AGG, b2, N);

    // ---- mean pool ----
    pool_clear <<<(NGRAPH * D + T - 1) / T, T, 0, stream>>>(pool, cnt);
    pool_sum   <<<gNwave, T, 0, stream>>>(AGG, batch, pool, N);
    pool_count <<<gN, T, 0, stream>>>(batch, cnt, N);
    pool_mean  <<<(NGRAPH * D + T - 1) / T, T, 0, stream>>>(pool, cnt);

    // ---- MLP head ----
    gemm_f32_wmma <<<gGemmG, T, 0, stream>>>(pool, lin1W, g1, NGRAPH);
    bias_relu     <<<(NGRAPH * D + T - 1) / T, T, 0, stream>>>(g1, lin1b, NGRAPH);
    final_lin     <<<(NGRAPH + T - 1) / T, T, 0, stream>>>(g1, lin2W, lin2b, out);
}